// YoloLoss_58609123721466
// MI455X (gfx1250) — compile-verified
//
#include <hip/hip_runtime.h>
#include <math.h>

// ---------------- problem constants (match reference) ----------------
static constexpr int kB   = 16;
static constexpr int kA   = 3;
static constexpr int kH   = 80;
static constexpr int kW   = 80;
static constexpr int kNC  = 80;
static constexpr int kN   = 2048;
static constexpr int kCH  = 5 + kNC;               // 85 channels
static constexpr int kBAHW = kB * kA * kH * kW;    // 307200
static constexpr float kEPS = 1e-7f;

// obj sweep geometry: 307200 = 600 blocks * 512 threads exactly
static constexpr int kObjBlk   = 512;
static constexpr int kObjGrid  = kBAHW / kObjBlk;  // 600
// target kernel: 2048 = 8 * 256 exactly
static constexpr int kTgtBlk   = 256;
static constexpr int kTgtGrid  = kN / kTgtBlk;     // 8

typedef __attribute__((ext_vector_type(2))) float v2f;
typedef __attribute__((ext_vector_type(8))) float v8f;

// ---------------- math helpers ----------------
__device__ __forceinline__ float sigmoidf_(float x) { return 1.0f / (1.0f + __expf(-x)); }

// BCE-with-logits single term: max(x,0) - x*t + log1p(exp(-|x|))
__device__ __forceinline__ float bce_term(float x, float t) {
    return fmaxf(x, 0.0f) - x * t + log1pf(__expf(-fabsf(x)));
}

// ---------------- wave32 reduction via V_WMMA_F32_16X16X4_F32 ----------------
// A = ones(16x4); B carries lane partials.  32-bit B operand layout (ISA 7.12.2,
// mirrored from the 16x4 A layout): VGPR0 lanes0-15 = row K=0, lanes16-31 = K=2;
// VGPR1 lanes0-15 = K=1, lanes16-31 = K=3.  With A=ones:
//   D[m][n] = B[0][n]+B[1][n]+B[2][n]+B[3][n]   (same for every m)
// C/D layout: c[0], lanes 0-15 hold D[0][n] (n = lane), lanes 16-31 hold D[8][n]
// which is identical since all rows of D match -> a 4-step xor-shuffle over the
// 16-column sums finishes the 32-lane reduction.
__device__ __forceinline__ float wave_reduce_sum(float p) {
#if __has_builtin(__builtin_amdgcn_wmma_f32_16x16x4_f32)
    v2f a; a.x = 1.0f; a.y = 1.0f;     // A = ones
    v2f b; b.x = p;    b.y = 0.0f;     // rows K=0 (lanes<16) / K=2 (lanes>=16)
    v8f c = {};
    c = __builtin_amdgcn_wmma_f32_16x16x4_f32(false, a, false, b, (short)0, c,
                                              false, false);
    float s = c[0];                    // lane n: p_n + p_(n+16)  (both halves)
    s += __shfl_xor(s, 8, 32);
    s += __shfl_xor(s, 4, 32);
    s += __shfl_xor(s, 2, 32);
    s += __shfl_xor(s, 1, 32);
    return s;                          // all lanes hold the 32-lane sum
#else
    for (int o = 16; o > 0; o >>= 1) p += __shfl_xor(p, o, 32);
    return p;
#endif
}

// ---------------- kernel 0: zero tobj scratch ----------------
__global__ void yolo_zero_tobj(float* __restrict__ tobj) {
    int i = blockIdx.x * blockDim.x + threadIdx.x;
    if (i < kBAHW) tobj[i] = 0.0f;
}

// ---------------- kernel 1: per-target CIoU + scatter + lcls ----------------
// One thread per target. Writes clip(ciou,0) into dense tobj, and per-block
// deterministic LDS-tree partial sums of (1-ciou) and class-BCE.
__global__ void yolo_targets(const float* __restrict__ pout,
                             const float* __restrict__ tbox,
                             const float* __restrict__ tanchors,
                             const int*   __restrict__ batch_id,
                             const int*   __restrict__ anc_id,
                             const int*   __restrict__ gy,
                             const int*   __restrict__ gx,
                             const int*   __restrict__ tcls,
                             float* __restrict__ tobj,
                             float* __restrict__ pA /* [kTgtGrid*2] */) {
    __shared__ float s_box[kTgtBlk];
    __shared__ float s_cls[kTgtBlk];

    const int i = blockIdx.x * blockDim.x + threadIdx.x;   // exact: 8*256=2048

    const int flat = ((batch_id[i] * kA + anc_id[i]) * kH + gy[i]) * kW + gx[i];
    const float* ps = pout + (size_t)flat * kCH;

    // predicted box
    const float px = sigmoidf_(ps[0]);
    const float py = sigmoidf_(ps[1]);
    const float pw = __expf(ps[2]) * tanchors[2 * i + 0];
    const float ph = __expf(ps[3]) * tanchors[2 * i + 1];
    // target box
    const float tx = tbox[4 * i + 0], ty = tbox[4 * i + 1];
    const float tw = tbox[4 * i + 2], th = tbox[4 * i + 3];

    const float b1x1 = px - pw * 0.5f, b1x2 = px + pw * 0.5f;
    const float b1y1 = py - ph * 0.5f, b1y2 = py + ph * 0.5f;
    const float b2x1 = tx - tw * 0.5f, b2x2 = tx + tw * 0.5f;
    const float b2y1 = ty - th * 0.5f, b2y2 = ty + th * 0.5f;

    const float iw = fmaxf(fminf(b1x2, b2x2) - fmaxf(b1x1, b2x1), 0.0f);
    const float ih = fmaxf(fminf(b1y2, b2y2) - fmaxf(b1y1, b2y1), 0.0f);
    const float inter = iw * ih;
    const float uni   = pw * ph + tw * th - inter + kEPS;
    const float iou   = inter / uni;

    const float cw = fmaxf(b1x2, b2x2) - fminf(b1x1, b2x1);
    const float ch = fmaxf(b1y2, b2y2) - fminf(b1y1, b2y1);
    const float c2 = cw * cw + ch * ch + kEPS;
    const float dx = b2x1 + b2x2 - b1x1 - b1x2;
    const float dy = b2y1 + b2y2 - b1y1 - b1y2;
    const float rho2 = (dx * dx + dy * dy) * 0.25f;

    const float dAtan = atanf(tw / (th + kEPS)) - atanf(pw / (ph + kEPS));
    const float v = (4.0f / (float)(M_PI * M_PI)) * dAtan * dAtan;
    const float alpha = v / (v - iou + (1.0f + kEPS));   // stop_gradient: fwd value
    const float ciou = iou - (rho2 / c2 + v * alpha);

    // scatter objectness target (dense)
    tobj[flat] = fmaxf(ciou, 0.0f);

    // class BCE over 80 logits vs one-hot
    const int tc = tcls[i];
    float cls_sum = 0.0f;
    #pragma unroll 8
    for (int c = 0; c < kNC; ++c) {
        const float x = ps[5 + c];
        cls_sum += bce_term(x, (c == tc) ? 1.0f : 0.0f);
    }

    // deterministic block tree reduce
    s_box[threadIdx.x] = 1.0f - ciou;
    s_cls[threadIdx.x] = cls_sum;
    __syncthreads();
    for (int off = kTgtBlk / 2; off > 0; off >>= 1) {
        if (threadIdx.x < off) {
            s_box[threadIdx.x] += s_box[threadIdx.x + off];
            s_cls[threadIdx.x] += s_cls[threadIdx.x + off];
        }
        __syncthreads();
    }
    if (threadIdx.x == 0) {
        pA[2 * blockIdx.x + 0] = s_box[0];
        pA[2 * blockIdx.x + 1] = s_cls[0];
    }
}

// ---------------- kernel 2: obj BCE sweep (the bandwidth-bound part) ----------
__global__ void yolo_obj(const float* __restrict__ pout,
                         const float* __restrict__ tobj,
                         float* __restrict__ pB /* [kObjGrid] */) {
    __shared__ float s_wave[kObjBlk / 32];   // 16 waves

    const int i = blockIdx.x * blockDim.x + threadIdx.x;  // exact coverage
    const float x = pout[(size_t)i * kCH + 4];
    const float t = tobj[i];
    float p = bce_term(x, t);

    // wave-level reduce through WMMA (full EXEC: 512 threads, no divergence)
    p = wave_reduce_sum(p);

    const int lane = threadIdx.x & 31;
    const int wave = threadIdx.x >> 5;
    if (lane == 0) s_wave[wave] = p;
    __syncthreads();

    if (wave == 0) {
        float s = (lane < (kObjBlk / 32)) ? s_wave[lane] : 0.0f;
        for (int o = 8; o > 0; o >>= 1) s += __shfl_xor(s, o, 32);
        if (lane == 0) pB[blockIdx.x] = s;
    }
}

// ---------------- kernel 3: final combine (1 wave) ----------------
// Accumulating WMMA reduction over the 600 obj-block partials:
//   C <- ones(16x4) x B(4x16) + C  folds 64 partials per v_wmma.
__global__ void yolo_final(const float* __restrict__ pA,
                           const float* __restrict__ pB,
                           float* __restrict__ out) {
    const int lane = threadIdx.x;   // 32 threads, all active

    float obj_sum;
#if __has_builtin(__builtin_amdgcn_wmma_f32_16x16x4_f32)
    v2f a; a.x = 1.0f; a.y = 1.0f;
    v8f c = {};
    const int chunks = (kObjGrid + 63) / 64;   // 10, uniform
    for (int k = 0; k < chunks; ++k) {
        const int i0 = k * 64 + lane;
        const int i1 = k * 64 + 32 + lane;
        v2f b;
        b.x = (i0 < kObjGrid) ? pB[i0] : 0.0f;  // rows K=0 / K=2
        b.y = (i1 < kObjGrid) ? pB[i1] : 0.0f;  // rows K=1 / K=3
        c = __builtin_amdgcn_wmma_f32_16x16x4_f32(false, a, false, b, (short)0,
                                                  c, false, false);
    }
    float s = c[0];                 // lane n: column sum over all chunks
    s += __shfl_xor(s, 8, 32);
    s += __shfl_xor(s, 4, 32);
    s += __shfl_xor(s, 2, 32);
    s += __shfl_xor(s, 1, 32);
    obj_sum = s;
#else
    float s = 0.0f;
    for (int i = lane; i < kObjGrid; i += 32) s += pB[i];
    for (int o = 16; o > 0; o >>= 1) s += __shfl_xor(s, o, 32);
    obj_sum = s;
#endif

    if (lane == 0) {
        float box_sum = 0.0f, cls_sum = 0.0f;
        for (int k = 0; k < kTgtGrid; ++k) {
            box_sum += pA[2 * k + 0];
            cls_sum += pA[2 * k + 1];
        }
        const float lbox = box_sum / (float)kN;
        const float lcls = cls_sum / (float)(kN * kNC);
        const float lobj = obj_sum / (float)kBAHW;
        out[0] = 0.05f * lcls + 1.0f * lobj + 0.5f * lbox;
    }
}

// ---------------- launcher ----------------
extern "C" void kernel_launch(void* const* d_in, const int* in_sizes, int n_in,
                              void* d_out, int out_size, void* d_ws, size_t ws_size,
                              hipStream_t stream) {
    const float* pout     = (const float*)d_in[0];
    const float* tbox     = (const float*)d_in[1];
    const float* tanchors = (const float*)d_in[2];
    const int*   batch_id = (const int*)d_in[3];
    const int*   anc_id   = (const int*)d_in[4];
    const int*   gy       = (const int*)d_in[5];
    const int*   gx       = (const int*)d_in[6];
    const int*   tcls     = (const int*)d_in[7];

    // workspace layout: [tobj: 307200 f32][pA: 16 f32][pB: 600 f32]
    float* tobj = (float*)d_ws;
    float* pA   = tobj + kBAHW;
    float* pB   = pA + 2 * kTgtGrid;

    yolo_zero_tobj<<<kObjGrid, kObjBlk, 0, stream>>>(tobj);
    yolo_targets<<<kTgtGrid, kTgtBlk, 0, stream>>>(pout, tbox, tanchors,
                                                   batch_id, anc_id, gy, gx,
                                                   tcls, tobj, pA);
    yolo_obj<<<kObjGrid, kObjBlk, 0, stream>>>(pout, tobj, pB);
    yolo_final<<<1, 32, 0, stream>>>(pA, pB, (float*)d_out);
}